// HAGCN_80152679678284
// MI455X (gfx1250) — compile-verified
//
#include <hip/hip_runtime.h>
#include <hip/hip_bf16.h>
#include <math.h>

// ---------------------------------------------------------------------------
// HAGCN forward for MI455X (gfx1250, wave32, WMMA).
// Pipeline:
//   1) h   = x @ Wlin^T + b              (WMMA f32 GEMM, NT layout)
//   2) f1  = h @ a[:F], f2 = h @ a[F:]   (block-reduction GEMV)
//   3) deg = [adj.sum(1); adj.sum(0)], dinv = 1/sqrt(1+deg)
//   4) Tgcn row kernel: masked-softmax(lrelu(f1_i+f2_j)) * diag scalings
//   5) WT  = W^T, xw = x @ W (via WT), xwT = xw^T
//   6) out = relu(Tgcn @ xw)             (WMMA f32 GEMM, NT layout)
// All GEMMs use V_WMMA_F32_16X16X4_F32 (full fp32 precision, fp32 accum).
// ---------------------------------------------------------------------------

typedef float v2f __attribute__((ext_vector_type(2)));
typedef float v8f __attribute__((ext_vector_type(8)));

#define GEMM_FLAG_BIAS 1
#define GEMM_FLAG_RELU 2

// NT GEMM: C[M,Nd] = A[M,K] (row-major, K contig) * B^T where B is [Nd,K]
// (row-major, K contig).  Wave tile 64x64 (4x4 WMMA frags), WG = 8 waves in
// 2(M) x 4(N) arrangement -> 128x256 per workgroup.
__global__ __launch_bounds__(256, 1)
void gemm_nt_wmma_f32(const float* __restrict__ A,
                      const float* __restrict__ B,
                      const float* __restrict__ bias,
                      float* __restrict__ C,
                      int M, int Nd, int K, int flags)
{
    const int lane   = threadIdx.x & 31;
    const int wave   = threadIdx.x >> 5;   // 0..7
    const int lane16 = lane & 15;
    const int half   = lane >> 4;          // 0: K {0,1}, 1: K {2,3}

    const int waveM = wave >> 2;           // 0..1
    const int waveN = wave & 3;            // 0..3
    const int m0 = blockIdx.y * 128 + waveM * 64;
    const int n0 = blockIdx.x * 256 + waveN * 64;

    // Per-lane row pointers; fold the lane-half K offset (+2*half) in now.
    const float* arow[4];
    const float* brow[4];
#pragma unroll
    for (int i = 0; i < 4; ++i)
        arow[i] = A + (size_t)(m0 + i * 16 + lane16) * K + 2 * half;
#pragma unroll
    for (int j = 0; j < 4; ++j)
        brow[j] = B + (size_t)(n0 + j * 16 + lane16) * K + 2 * half;

    const v8f vzero = {};
    v8f acc[4][4];
#pragma unroll
    for (int i = 0; i < 4; ++i)
#pragma unroll
        for (int j = 0; j < 4; ++j)
            acc[i][j] = vzero;

    for (int k = 0; k < K; k += 4) {
        v2f afrag[4];
        v2f bfrag[4];
#pragma unroll
        for (int i = 0; i < 4; ++i)
            afrag[i] = *(const v2f*)(arow[i] + k);
#pragma unroll
        for (int j = 0; j < 4; ++j)
            bfrag[j] = *(const v2f*)(brow[j] + k);

#pragma unroll
        for (int i = 0; i < 4; ++i)
#pragma unroll
            for (int j = 0; j < 4; ++j)
                acc[i][j] = __builtin_amdgcn_wmma_f32_16x16x4_f32(
                    /*neg_a=*/false, afrag[i],
                    /*neg_b=*/false, bfrag[j],
                    /*c_mod=*/(short)0, acc[i][j],
                    /*reuse_a=*/false, /*reuse_b=*/false);
    }

    // Epilogue. C/D layout: VGPR v -> row (v + 8*half), lane16 -> col.
#pragma unroll
    for (int j = 0; j < 4; ++j) {
        const int col = n0 + j * 16 + lane16;
        const float bc = (flags & GEMM_FLAG_BIAS) ? bias[col] : 0.0f;
#pragma unroll
        for (int i = 0; i < 4; ++i) {
#pragma unroll
            for (int v = 0; v < 8; ++v) {
                float val = acc[i][j][v] + bc;
                if (flags & GEMM_FLAG_RELU) val = fmaxf(val, 0.0f);
                const int row = m0 + i * 16 + v + 8 * half;
                C[(size_t)row * Nd + col] = val;
            }
        }
    }
}

// f1[i] = sum_o h[i,o]*a[o]; f2[i] = sum_o h[i,o]*a[F+o]. One block per row.
__global__ __launch_bounds__(256)
void fvec_kernel(const float* __restrict__ h, const float* __restrict__ a,
                 float* __restrict__ f1, float* __restrict__ f2, int F)
{
    __shared__ float s1[256];
    __shared__ float s2[256];
    const int row = blockIdx.x;
    const int tid = threadIdx.x;
    const float* hr = h + (size_t)row * F;
    float p1 = 0.0f, p2 = 0.0f;
    for (int o = tid; o < F; o += 256) {
        const float hv = hr[o];
        p1 += hv * a[o];
        p2 += hv * a[F + o];
    }
    s1[tid] = p1; s2[tid] = p2;
    __syncthreads();
    for (int s = 128; s > 0; s >>= 1) {
        if (tid < s) { s1[tid] += s1[tid + s]; s2[tid] += s2[tid + s]; }
        __syncthreads();
    }
    if (tid == 0) { f1[row] = s1[0]; f2[row] = s2[0]; }
}

// dinv[i] = 1/sqrt(1 + degree[i]); degree = [adj row sums ; adj col sums].
__global__ __launch_bounds__(256)
void degree_kernel(const float* __restrict__ adj, float* __restrict__ dinv,
                   int hn)
{
    __shared__ float ss[256];
    const int idx = blockIdx.x;     // 0 .. 2*hn-1
    const int tid = threadIdx.x;
    float p = 0.0f;
    if (idx < hn) {
        const float* r = adj + (size_t)idx * hn;
        for (int c = tid; c < hn; c += 256) p += r[c];
    } else {
        const int c = idx - hn;
        for (int r = tid; r < hn; r += 256) p += adj[(size_t)r * hn + c];
    }
    ss[tid] = p;
    __syncthreads();
    for (int s = 128; s > 0; s >>= 1) {
        if (tid < s) ss[tid] += ss[tid + s];
        __syncthreads();
    }
    if (tid == 0) dinv[idx] = 1.0f / sqrtf(1.0f + ss[0]);
}

// Per row i: e_j = lrelu(f1[i]+f2[j]); v_j = x[i,j]>0 ? e_j : -9e15;
// att = softmax(v); Tg[i,j] = dinv[i]*ddinv*att_j*dinv[j].
// (att.sum(1) == 1 after softmax, so ddinv = 1/(1+1e-6).)
__global__ __launch_bounds__(256)
void attn_kernel(const float* __restrict__ x, const float* __restrict__ f1,
                 const float* __restrict__ f2, const float* __restrict__ dinv,
                 float* __restrict__ Tg, int n)
{
    __shared__ float red[256];
    const int i = blockIdx.x;
    const int tid = threadIdx.x;
    const float* xr = x + (size_t)i * n;
    const float f1i = f1[i];

    // pass 1: row max of masked logits
    float m = -INFINITY;
    for (int j = tid; j < n; j += 256) {
        float e = f1i + f2[j];
        e = (e > 0.0f) ? e : 0.01f * e;
        const float v = (xr[j] > 0.0f) ? e : -9e15f;
        m = fmaxf(m, v);
    }
    red[tid] = m;
    __syncthreads();
    for (int s = 128; s > 0; s >>= 1) {
        if (tid < s) red[tid] = fmaxf(red[tid], red[tid + s]);
        __syncthreads();
    }
    m = red[0];
    __syncthreads();

    // pass 2: sum of exp
    float z = 0.0f;
    for (int j = tid; j < n; j += 256) {
        float e = f1i + f2[j];
        e = (e > 0.0f) ? e : 0.01f * e;
        const float v = (xr[j] > 0.0f) ? e : -9e15f;
        z += __expf(v - m);
    }
    red[tid] = z;
    __syncthreads();
    for (int s = 128; s > 0; s >>= 1) {
        if (tid < s) red[tid] += red[tid + s];
        __syncthreads();
    }
    z = red[0];
    __syncthreads();

    const float invz  = 1.0f / z;
    const float ddinv = 1.0f / (1.0f + 1e-6f);   // att row-sum == 1
    const float rs    = dinv[i] * ddinv * invz;

    // pass 3: write Tgcn row
    float* tr = Tg + (size_t)i * n;
    for (int j = tid; j < n; j += 256) {
        float e = f1i + f2[j];
        e = (e > 0.0f) ? e : 0.01f * e;
        const float v = (xr[j] > 0.0f) ? e : -9e15f;
        tr[j] = rs * __expf(v - m) * dinv[j];
    }
}

// out[cols, rows] = in[rows, cols]^T.  32x32 LDS tiles, 256 threads.
__global__ __launch_bounds__(256)
void transpose_kernel(const float* __restrict__ in, float* __restrict__ out,
                      int rows, int cols)
{
    __shared__ float tile[32][33];
    const int bx = blockIdx.x * 32;   // col base in `in`
    const int by = blockIdx.y * 32;   // row base in `in`
    const int tx = threadIdx.x & 31;
    const int ty = threadIdx.x >> 5;  // 0..7
#pragma unroll
    for (int yy = ty; yy < 32; yy += 8)
        tile[yy][tx] = in[(size_t)(by + yy) * cols + bx + tx];
    __syncthreads();
#pragma unroll
    for (int yy = ty; yy < 32; yy += 8)
        out[(size_t)(bx + yy) * rows + by + tx] = tile[tx][yy];
}

extern "C" void kernel_launch(void* const* d_in, const int* in_sizes, int n_in,
                              void* d_out, int out_size, void* d_ws, size_t ws_size,
                              hipStream_t stream)
{
    (void)in_sizes; (void)n_in; (void)out_size; (void)ws_size;

    const int N    = 4096;
    const int OUTF = 1024;
    const int HN   = 2048;

    const float* x    = (const float*)d_in[0];   // [N, N]
    const float* adj  = (const float*)d_in[1];   // [HN, HN]
    const float* Wlin = (const float*)d_in[2];   // [OUTF, N]  (K-contig already)
    const float* b    = (const float*)d_in[3];   // [OUTF]
    const float* a    = (const float*)d_in[4];   // [2*OUTF]
    const float* W    = (const float*)d_in[5];   // [N, OUTF]
    float* out = (float*)d_out;                  // [N, OUTF]

    // Workspace layout.
    char* ws = (char*)d_ws;
    size_t off = 0;
    float* Tg  = (float*)(ws + off); off += (size_t)N * N * sizeof(float);     // 64MB
    float* hb  = (float*)(ws + off); off += (size_t)N * OUTF * sizeof(float);  // 16MB (h, then W^T)
    float* xw  = (float*)(ws + off); off += (size_t)N * OUTF * sizeof(float);  // 16MB
    float* xwT = (float*)(ws + off); off += (size_t)N * OUTF * sizeof(float);  // 16MB
    float* f1   = (float*)(ws + off); off += (size_t)N * sizeof(float);
    float* f2   = (float*)(ws + off); off += (size_t)N * sizeof(float);
    float* dinv = (float*)(ws + off); off += (size_t)N * sizeof(float);

    const dim3 gemm_grid(OUTF / 256, N / 128);
    const dim3 tr_grid(OUTF / 32, N / 32);

    // 1) h = x @ Wlin^T + b
    gemm_nt_wmma_f32<<<gemm_grid, 256, 0, stream>>>(x, Wlin, b, hb,
                                                    N, OUTF, N, GEMM_FLAG_BIAS);
    // 2) f1, f2
    fvec_kernel<<<N, 256, 0, stream>>>(hb, a, f1, f2, OUTF);
    // 3) degree -> dinv
    degree_kernel<<<N, 256, 0, stream>>>(adj, dinv, HN);
    // 4) Tgcn
    attn_kernel<<<N, 256, 0, stream>>>(x, f1, f2, dinv, Tg, N);
    // 5) WT = W^T (reuse h buffer), xw = x @ W, xwT = xw^T
    float* WT = hb;
    transpose_kernel<<<tr_grid, 256, 0, stream>>>(W, WT, N, OUTF);
    gemm_nt_wmma_f32<<<gemm_grid, 256, 0, stream>>>(x, WT, nullptr, xw,
                                                    N, OUTF, N, 0);
    transpose_kernel<<<tr_grid, 256, 0, stream>>>(xw, xwT, N, OUTF);
    // 6) out = relu(Tgcn @ xw)
    gemm_nt_wmma_f32<<<gemm_grid, 256, 0, stream>>>(Tg, xwT, nullptr, out,
                                                    N, OUTF, N, GEMM_FLAG_RELU);
}